// MoEAlltoAllTokenDispatcher_22162031247685
// MI455X (gfx1250) — compile-verified
//
#include <hip/hip_runtime.h>
#include <stdint.h>

// MoE dispatch round-trip, collapsed:
//   restored[t,:]        = hidden[t,:] * sum_e probs[t,e]
//   tokens_per_expert[e] = count_t (probs[t,e] > 0)
//
// Bandwidth-bound (~256 MiB @ 23.3 TB/s ~= 11.5 us). Uses CDNA5 async
// global->LDS b128 copies (ASYNCcnt) + b128 non-temporal stores + wave32
// ballot. No WMMA: a diagonal scale has zero matrix reuse.

#define T_TOKENS 8192
#define HDIM     4096
#define NEXP     8

typedef float v4f __attribute__((ext_vector_type(4)));

__global__ __launch_bounds__(256) void moe_scale_rows(
    const float* __restrict__ hidden,
    const float* __restrict__ probs,
    float* __restrict__ out) {
  __shared__ v4f tile[HDIM / 4];   // 16 KB row staging buffer
  __shared__ float s_scale;

  const int t   = blockIdx.x;      // token row
  const int tid = threadIdx.x;

  // Router weight sum for this token (8 floats -> 2x b128 loads by lane 0).
  if (tid == 0) {
    const v4f* p4 = (const v4f*)(probs + (size_t)t * NEXP);
    v4f a = p4[0];
    v4f b = p4[1];
    s_scale = ((a.x + a.y) + (a.z + a.w)) + ((b.x + b.y) + (b.z + b.w));
  }

  // Stage the 16 KB hidden row into LDS with async 128-bit copies.
  // Each of 256 lanes issues 4 copies: covers 1024 x float4 = 4096 floats.
  const char* grow = (const char*)(hidden + (size_t)t * HDIM);
#pragma unroll
  for (int c = 0; c < 4; ++c) {
    const int f4 = c * 256 + tid;  // float4 index within the row
    const unsigned lds_addr =
        (unsigned)(uintptr_t)&tile[f4];
    const unsigned long long gaddr =
        (unsigned long long)(uintptr_t)(grow + (size_t)f4 * 16);
    asm volatile("global_load_async_to_lds_b128 %0, %1, off"
                 :: "v"(lds_addr), "v"(gaddr)
                 : "memory");
  }
  // Wait for this wave's async LDS writes (ASYNCcnt), then publish s_scale.
  asm volatile("s_wait_asynccnt 0x0" ::: "memory");
  __syncthreads();

  const float scale = s_scale;
  v4f* __restrict__ o4 = (v4f*)(out + (size_t)t * HDIM);
#pragma unroll
  for (int c = 0; c < 4; ++c) {
    const int f4 = c * 256 + tid;
    v4f v = tile[f4];              // ds_load_b128
    v.x *= scale; v.y *= scale; v.z *= scale; v.w *= scale;
    __builtin_nontemporal_store(v, &o4[f4]);  // write-once: NT b128 store
  }
}

__global__ __launch_bounds__(1024) void moe_count_tokens(
    const float* __restrict__ probs,
    int* __restrict__ counts) {
  __shared__ int scnt[NEXP];
  const int tid  = threadIdx.x;
  const unsigned lane = tid & 31u;           // wave32

  if (tid < NEXP) scnt[tid] = 0;
  __syncthreads();

  unsigned local[NEXP] = {0, 0, 0, 0, 0, 0, 0, 0};
  for (int t = tid; t < T_TOKENS; t += 1024) {
    const v4f* p4 = (const v4f*)(probs + (size_t)t * NEXP);
    v4f a = p4[0];
    v4f b = p4[1];
    float p[NEXP] = {a.x, a.y, a.z, a.w, b.x, b.y, b.z, b.w};
#pragma unroll
    for (int e = 0; e < NEXP; ++e) {
      unsigned long long m = __ballot(p[e] > 0.0f);   // wave32 mask
      if (lane == 0) local[e] += (unsigned)__popcll(m);
    }
  }
  if (lane == 0) {
#pragma unroll
    for (int e = 0; e < NEXP; ++e)
      if (local[e]) atomicAdd(&scnt[e], (int)local[e]);
  }
  __syncthreads();
  if (tid < NEXP) counts[tid] = scnt[tid];
}

extern "C" void kernel_launch(void* const* d_in, const int* in_sizes, int n_in,
                              void* d_out, int out_size, void* d_ws, size_t ws_size,
                              hipStream_t stream) {
  (void)in_sizes; (void)n_in; (void)d_ws; (void)ws_size; (void)out_size;

  const float* hidden = (const float*)d_in[0];   // [T, H] float32
  const float* probs  = (const float*)d_in[1];   // [T, E] float32 (zeroed off-route)
  // d_in[2] (routing_map bool) and d_in[3] (top_k) are not needed.

  float* restored = (float*)d_out;                              // [T, H] float32
  int*   counts   = (int*)(restored + (size_t)T_TOKENS * HDIM); // [E] int32 bits

  moe_scale_rows<<<T_TOKENS, 256, 0, stream>>>(hidden, probs, restored);
  moe_count_tokens<<<1, 1024, 0, stream>>>(probs, counts);
}